// GARCH_15006615733132
// MI455X (gfx1250) — compile-verified
//
#include <hip/hip_runtime.h>
#include <math.h>

typedef float v2f __attribute__((ext_vector_type(2)));
typedef float v8f __attribute__((ext_vector_type(8)));

#define GARCH_CHUNK   2048      // b-elements per chunk (one wave32 per chunk)
#define GARCH_GROUPS  8         // 2048 / 256 elements per WMMA group
#define SCAN_THREADS  1024
#define MAX_CHUNKS    8192      // ceil((2^24 - 1) / 2048)

#if defined(__AMDGCN__) && __has_builtin(__builtin_amdgcn_wmma_f32_16x16x4_f32)
#define GARCH_HAVE_WMMA 1
#else
#define GARCH_HAVE_WMMA 0
#endif

#if defined(__AMDGCN__) && \
    __has_builtin(__builtin_amdgcn_global_load_async_to_lds_b128) && \
    __has_builtin(__builtin_amdgcn_s_wait_asynccnt)
#define GARCH_HAVE_ASYNC 1
#else
#define GARCH_HAVE_ASYNC 0
#endif

// wave32 lane gather (deterministic, uses DS_BPERMUTE_B32; index wraps mod 32)
static __device__ __forceinline__ float lane_get(float v, int src) {
  return __builtin_bit_cast(
      float, __builtin_amdgcn_ds_bpermute(src << 2, __builtin_bit_cast(int, v)));
}

static __device__ __forceinline__ float powi_f(float b, int e) {
  float r = 1.0f;
  for (int i = 0; i < e; ++i) r *= b;
  return r;
}

// b^(2^k) by repeated squaring
static __device__ __forceinline__ float pow2k_f(float b, int k) {
  float r = b;
  for (int i = 0; i < k; ++i) r *= r;
  return r;
}

// Stage one 2048-float tile of `ret` into LDS (coalesced; zero-pad past n).
// Uses the CDNA5 async memory->LDS path when available (ASYNCcnt), else
// classic load+ds_store.
static __device__ __forceinline__ void garch_stage_lds(
    const float* __restrict__ ret, float* lds, size_t rbase, int n, int lane) {
#if GARCH_HAVE_ASYNC
  if (rbase + GARCH_CHUNK <= (size_t)n) {
    // Builtin signature (from compiler diagnostic): param0 is
    // 'int __attribute__((vector_size(16))) __device__ *' (4xi32, AS1);
    // param1 is the LDS-side counterpart (AS3).
    typedef int gar_v4i __attribute__((vector_size(16)));
    typedef gar_v4i __attribute__((address_space(1)))* g128p;
    typedef gar_v4i __attribute__((address_space(3)))* l128p;
    g128p g = (g128p)(ret + rbase);
    l128p l = (l128p)lds;
#pragma unroll
    for (int it = 0; it < GARCH_CHUNK / 128; ++it) {
      const int e = lane + 32 * it;   // 16-byte element index
      __builtin_amdgcn_global_load_async_to_lds_b128(g + e, l + e, 0, 0);
    }
    __builtin_amdgcn_s_wait_asynccnt(0);
    return;
  }
#endif
  if (rbase + GARCH_CHUNK <= (size_t)n) {
    const float4* g4 = (const float4*)(ret + rbase);
    float4* l4 = (float4*)lds;
#pragma unroll
    for (int it = 0; it < GARCH_CHUNK / 128; ++it)
      l4[lane + 32 * it] = g4[lane + 32 * it];
  } else {
    for (int i = lane; i < GARCH_CHUNK; i += 32) {
      size_t ri = rbase + i;
      lds[i] = (ri < (size_t)n) ? ret[ri] : 0.0f;
    }
  }
}

// Affine scan of one 2048-element chunk held in LDS (as raw returns r).
// h[t] = (omega + alpha*r^2) + beta*h[t-1], seeded with h_in.
// If EMIT, per-element h overwrites the LDS tile in place.
// Returns h at the end of the chunk (uniform across lanes).
//
// Core: per 256-element group, D = T * B via 4 chained V_WMMA_F32_16X16X4_F32,
// T = 16x16 lower-triangular Toeplitz of beta powers, B columns = 16 segments.
template <bool EMIT>
static __device__ __forceinline__ float garch_chunk_scan(
    float* lds, float omega, float alpha, float beta, float h_in, int lane) {
  const int col = lane & 15;   // column n (B/D) or row m (A) depending on role
  const int kh  = lane >> 4;   // lane-half selector (high K / high M half)

  // A operand tiles: A_c = T[:, 4c .. 4c+3], ISA 32-bit 16x4 layout:
  // VGPR0 = K even-pair base, VGPR1 = +1; lane halves split K by 2.
  v2f Aop[4];
#pragma unroll
  for (int c = 0; c < 4; ++c) {
    int k0 = 4 * c + 2 * kh;
    Aop[c].x = (col >= k0)     ? powi_f(beta, col - k0)     : 0.0f;
    Aop[c].y = (col >= k0 + 1) ? powi_f(beta, col - k0 - 1) : 0.0f;
  }
  float fac[8];
#pragma unroll
  for (int rr = 0; rr < 8; ++rr) fac[rr] = powi_f(beta, rr + 8 * kh + 1);
  const float p16 = powi_f(beta, 16);

  float h = h_in;
  for (int g = 0; g < GARCH_GROUPS; ++g) {
    const int gbase = g * 256;
    v8f acc = {0.f, 0.f, 0.f, 0.f, 0.f, 0.f, 0.f, 0.f};
#if GARCH_HAVE_WMMA
#pragma unroll
    for (int c = 0; c < 4; ++c) {
      // B_c[k][j] = b[16*j + 4c + k]; lane holds (j=col, k-half=kh), VGPR=k&1
      const float2 rr2 = *(const float2*)(lds + gbase + 16 * col + 4 * c + 2 * kh);
      v2f B;
      B.x = omega + alpha * rr2.x * rr2.x;
      B.y = omega + alpha * rr2.y * rr2.y;
      acc = __builtin_amdgcn_wmma_f32_16x16x4_f32(
          false, Aop[c], false, B, (short)0, acc, false, false);
    }
#else
    // Scalar fallback: same math, per-lane (keeps compile green if builtin absent)
    float bcol[16];
#pragma unroll
    for (int k = 0; k < 16; ++k) {
      float rv = lds[gbase + 16 * col + k];
      bcol[k] = omega + alpha * rv * rv;
    }
#pragma unroll
    for (int rr = 0; rr < 8; ++rr) {
      int m = rr + 8 * kh;
      float a = 0.0f;
      for (int k = 0; k <= m; ++k) a += powi_f(beta, m - k) * bcol[k];
      acc[rr] = a;
    }
#endif
    // Segment composites: S_j = D[15][j] lives in VGPR7, lanes 16..31
    float S = lane_get(acc[7], 16 + col);
    // Inclusive Hillis-Steele over 16 segments, pairs (s, p), p starts beta^16.
    float s = S, p = p16;
#pragma unroll
    for (int off = 1; off <= 8; off <<= 1) {
      float s2 = lane_get(s, lane - off);
      float p2 = lane_get(p, lane - off);
      if (col >= off) { s = s + p * s2; p = p * p2; }
    }
    // Exclusive seed for segment col, applied to incoming h
    float se = lane_get(s, lane - 1);
    float pe = lane_get(p, lane - 1);
    if (col == 0) { se = 0.0f; pe = 1.0f; }
    const float seed = se + pe * h;
    const float sl = lane_get(s, 15);   // full-group composite (uniform)
    const float pl = lane_get(p, 15);   // beta^256
    if (EMIT) {
#pragma unroll
      for (int rr = 0; rr < 8; ++rr) {
        // D[m][n] -> element index 16*n + m, m = rr + 8*kh, n = col
        lds[gbase + 16 * col + rr + 8 * kh] = acc[rr] + fac[rr] * seed;
      }
    }
    h = sl + pl * h;   // chain into next group (uniform across lanes)
  }
  return h;
}

// ---- Kernel 1: per-chunk partial sums + chunk scan composite -------------
__global__ void __launch_bounds__(32) garch_stage1(
    const float* __restrict__ ret,
    const float* __restrict__ p_omega, const float* __restrict__ p_alpha,
    const float* __restrict__ p_beta,
    float* __restrict__ wsS, float* __restrict__ wsSum, float* __restrict__ wsSsq,
    int n) {
  __shared__ __align__(16) float lds[GARCH_CHUNK];
  const int lane = threadIdx.x;
  const int c = blockIdx.x;
  const size_t rbase = (size_t)c * GARCH_CHUNK;
  const float omega = p_omega[0], alpha = p_alpha[0], beta = p_beta[0];

  garch_stage_lds(ret, lds, rbase, n, lane);
  __syncthreads();

  float s = 0.0f, q = 0.0f;
  for (int i = lane; i < GARCH_CHUNK; i += 32) {
    float v = lds[i];
    s += v;
    q += v * v;
  }
#pragma unroll
  for (int off = 16; off >= 1; off >>= 1) {
    s += lane_get(s, lane ^ off);
    q += lane_get(q, lane ^ off);
  }

  const float S = garch_chunk_scan<false>(lds, omega, alpha, beta, 0.0f, lane);
  if (lane == 0) {
    wsS[c] = S;
    wsSum[c] = s;
    wsSsq[c] = q;
  }
}

// ---- Kernel 2: h0 (ddof=1 variance) + scan of chunk composites -----------
__global__ void __launch_bounds__(SCAN_THREADS) garch_stage2(
    const float* __restrict__ ret,
    const float* __restrict__ wsS, const float* __restrict__ wsSum,
    const float* __restrict__ wsSsq,
    float* __restrict__ carry,
    const float* __restrict__ p_beta,
    float* __restrict__ out, int n, int nChunks) {
  __shared__ float red[SCAN_THREADS];
  __shared__ float sArr[SCAN_THREADS];
  __shared__ float pArr[SCAN_THREADS];
  __shared__ float h0sh;
  const int t = threadIdx.x;
  const float beta = p_beta[0];

  float s = 0.0f, q = 0.0f;
  for (int j = t; j < nChunks; j += SCAN_THREADS) { s += wsSum[j]; q += wsSsq[j]; }
  red[t] = s;
  __syncthreads();
  for (int off = SCAN_THREADS / 2; off >= 1; off >>= 1) {
    if (t < off) red[t] += red[t + off];
    __syncthreads();
  }
  float sumTot = red[0];
  __syncthreads();
  red[t] = q;
  __syncthreads();
  for (int off = SCAN_THREADS / 2; off >= 1; off >>= 1) {
    if (t < off) red[t] += red[t + off];
    __syncthreads();
  }
  if (t == 0) {
    float ssqTot = red[0];
    // generic tail (chunks cover r[0 .. 2048*nChunks); add any remainder)
    for (size_t ri = (size_t)nChunks * GARCH_CHUNK; ri < (size_t)n; ++ri) {
      float v = ret[ri];
      sumTot += v;
      ssqTot += v * v;
    }
    float mean = sumTot / (float)n;
    float h0 = (ssqTot - mean * sumTot) / (float)(n - 1);
    h0sh = h0;
    out[0] = sqrtf(h0);   // sqrt(h) output, t = 0
    out[n] = h0;          // h output, t = 0
  }
  __syncthreads();
  const float h0 = h0sh;

  const float P = pow2k_f(beta, 11);  // beta^2048 (underflows to 0: exact)
  float Sv[8];
  float sc = 0.0f, pc = 1.0f;
  const int base = t * 8;
#pragma unroll
  for (int j = 0; j < 8; ++j) {
    int cidx = base + j;
    float Sj = (cidx < nChunks) ? wsS[cidx] : 0.0f;
    Sv[j] = Sj;
    if (cidx < nChunks) { sc = Sj + P * sc; pc = pc * P; }
  }
  sArr[t] = sc;
  pArr[t] = pc;
  __syncthreads();
  for (int off = 1; off < SCAN_THREADS; off <<= 1) {
    float s2 = 0.0f, p2 = 1.0f;
    if (t >= off) { s2 = sArr[t - off]; p2 = pArr[t - off]; }
    __syncthreads();
    if (t >= off) { sc = sc + pc * s2; pc = pc * p2; }
    sArr[t] = sc;
    pArr[t] = pc;
    __syncthreads();
  }
  float cs = (t == 0) ? h0 : (sArr[t - 1] + pArr[t - 1] * h0);
#pragma unroll
  for (int j = 0; j < 8; ++j) {
    int cidx = base + j;
    if (cidx < nChunks) {
      carry[cidx] = cs;             // h entering chunk cidx (== h[2048*cidx])
      cs = Sv[j] + P * cs;
    }
  }
}

// ---- Kernel 3: seeded per-chunk scan, emit sqrt(h) and h -----------------
__global__ void __launch_bounds__(32) garch_stage3(
    const float* __restrict__ ret,
    const float* __restrict__ carry,
    const float* __restrict__ p_omega, const float* __restrict__ p_alpha,
    const float* __restrict__ p_beta,
    float* __restrict__ out, int n) {
  __shared__ __align__(16) float lds[GARCH_CHUNK];
  const int lane = threadIdx.x;
  const int c = blockIdx.x;
  const size_t rbase = (size_t)c * GARCH_CHUNK;
  const float omega = p_omega[0], alpha = p_alpha[0], beta = p_beta[0];

  garch_stage_lds(ret, lds, rbase, n, lane);
  __syncthreads();

  garch_chunk_scan<true>(lds, omega, alpha, beta, carry[c], lane);
  __syncthreads();

  float* outS = out;
  float* outH = out + n;
  const size_t tbase = rbase + 1;   // chunk c covers t in [1 + 2048c, ...)
  for (int i = lane; i < GARCH_CHUNK; i += 32) {
    size_t tt = tbase + i;
    if (tt < (size_t)n) {
      float h = lds[i];
      // NT stores: don't let the 128MB output stream evict `ret` from L2
      __builtin_nontemporal_store(h, &outH[tt]);
      __builtin_nontemporal_store(sqrtf(h), &outS[tt]);
    }
  }
}

extern "C" void kernel_launch(void* const* d_in, const int* in_sizes, int n_in,
                              void* d_out, int out_size, void* d_ws, size_t ws_size,
                              hipStream_t stream) {
  const float* ret     = (const float*)d_in[0];
  const float* p_omega = (const float*)d_in[1];
  const float* p_alpha = (const float*)d_in[2];
  const float* p_beta  = (const float*)d_in[3];
  float* out = (float*)d_out;
  const int n = in_sizes[0];
  const int nChunks = (n - 1 + GARCH_CHUNK - 1) / GARCH_CHUNK;  // 8192 for N=2^24

  float* ws      = (float*)d_ws;
  float* wsS     = ws;                  // chunk composites
  float* wsSum   = ws + MAX_CHUNKS;     // chunk sum(r)
  float* wsSsq   = ws + 2 * MAX_CHUNKS; // chunk sum(r^2)
  float* wsCarry = ws + 3 * MAX_CHUNKS; // per-chunk entry carry

  garch_stage1<<<nChunks, 32, 0, stream>>>(ret, p_omega, p_alpha, p_beta,
                                           wsS, wsSum, wsSsq, n);
  garch_stage2<<<1, SCAN_THREADS, 0, stream>>>(ret, wsS, wsSum, wsSsq, wsCarry,
                                               p_beta, out, n, nChunks);
  garch_stage3<<<nChunks, 32, 0, stream>>>(ret, wsCarry, p_omega, p_alpha,
                                           p_beta, out, n);
}